// PointnetFPModule_40810779246764
// MI455X (gfx1250) — compile-verified
//
#include <hip/hip_runtime.h>
#include <hip/hip_bf16.h>

// Problem constants (match reference)
constexpr int kB  = 4;
constexpr int kN  = 8192;
constexpr int kM  = 2048;
constexpr int kC1 = 128;
constexpr int kC2 = 256;
constexpr int kCin = kC1 + kC2;   // 384
constexpr int kH1 = 256;
constexpr int kH2 = 128;
constexpr float kEpsDist = 1e-8f;
constexpr float kEpsBN   = 1e-5f;

typedef float v2f __attribute__((ext_vector_type(2)));
typedef float v8f __attribute__((ext_vector_type(8)));

// ---------------------------------------------------------------------------
// Kernel 1: 3-NN over known points + inverse-distance weights.
// One block per 256 unknown points of one batch; known coords staged in LDS.
// ---------------------------------------------------------------------------
__global__ __launch_bounds__(256) void knn_kernel(
    const float* __restrict__ unknown,   // (B, N, 3)
    const float* __restrict__ known,     // (B, M, 3)
    int*   __restrict__ idx_out,         // (B, N, 3)
    float* __restrict__ w_out)           // (B, N, 3)
{
    __shared__ float skx[kM];
    __shared__ float sky[kM];
    __shared__ float skz[kM];

    const int b = blockIdx.y;
    const int n = blockIdx.x * blockDim.x + threadIdx.x;

    const float* kb = known + (size_t)b * kM * 3;
    for (int i = threadIdx.x; i < kM; i += blockDim.x) {
        skx[i] = kb[i * 3 + 0];
        sky[i] = kb[i * 3 + 1];
        skz[i] = kb[i * 3 + 2];
    }
    __syncthreads();

    const float ux = unknown[((size_t)b * kN + n) * 3 + 0];
    const float uy = unknown[((size_t)b * kN + n) * 3 + 1];
    const float uz = unknown[((size_t)b * kN + n) * 3 + 2];

    float d0 = 3.4e38f, d1 = 3.4e38f, d2 = 3.4e38f;
    int   i0 = 0, i1 = 0, i2 = 0;

    for (int m = 0; m < kM; ++m) {
        const float dx = ux - skx[m];
        const float dy = uy - sky[m];
        const float dz = uz - skz[m];
        const float d  = dx * dx + dy * dy + dz * dz;
        if (d < d0) {
            d2 = d1; i2 = i1;
            d1 = d0; i1 = i0;
            d0 = d;  i0 = m;
        } else if (d < d1) {
            d2 = d1; i2 = i1;
            d1 = d;  i1 = m;
        } else if (d < d2) {
            d2 = d;  i2 = m;
        }
    }

    const float r0 = 1.0f / (d0 + kEpsDist);
    const float r1 = 1.0f / (d1 + kEpsDist);
    const float r2 = 1.0f / (d2 + kEpsDist);
    const float rs = 1.0f / (r0 + r1 + r2);

    const size_t base = ((size_t)b * kN + n) * 3;
    idx_out[base + 0] = i0;
    idx_out[base + 1] = i1;
    idx_out[base + 2] = i2;
    w_out[base + 0] = r0 * rs;
    w_out[base + 1] = r1 * rs;
    w_out[base + 2] = r2 * rs;
}

// ---------------------------------------------------------------------------
// Kernel 2: interpolate known_feats with 3-NN weights and concat unknow_feats
// into x (B, Cin=384, N).  Channel-major writes are coalesced across threads.
// ---------------------------------------------------------------------------
__global__ __launch_bounds__(256) void build_x_kernel(
    const float* __restrict__ known_feats,   // (B, C2, M)
    const float* __restrict__ unknow_feats,  // (B, C1, N)
    const int*   __restrict__ idx,           // (B, N, 3)
    const float* __restrict__ wgt,           // (B, N, 3)
    float* __restrict__ xbuf)                // (B, Cin, N)
{
    const int b = blockIdx.y;
    const int n = blockIdx.x * blockDim.x + threadIdx.x;

    const size_t base = ((size_t)b * kN + n) * 3;
    const int   i0 = idx[base + 0], i1 = idx[base + 1], i2 = idx[base + 2];
    const float w0 = wgt[base + 0], w1 = wgt[base + 1], w2 = wgt[base + 2];

    const float* kf = known_feats + (size_t)b * kC2 * kM;
    float* xb = xbuf + (size_t)b * kCin * kN;

    #pragma unroll 4
    for (int c = 0; c < kC2; ++c) {
        const float* row = kf + (size_t)c * kM;
        xb[(size_t)c * kN + n] = w0 * row[i0] + w1 * row[i1] + w2 * row[i2];
    }

    const float* uf = unknow_feats + (size_t)b * kC1 * kN;
    #pragma unroll 4
    for (int c = 0; c < kC1; ++c) {
        xb[(size_t)(kC2 + c) * kN + n] = uf[(size_t)c * kN + n];
    }
}

// ---------------------------------------------------------------------------
// Kernel 3: fp32 WMMA GEMM   Y[b,o,n] = sum_c W[o,c] * X[b,c,n]
// Each wave32 owns a 64(out-rows) x 16(cols) tile: 4 accumulators, K stepped
// by 4 with V_WMMA_F32_16X16X4_F32.  Lane layouts per CDNA5 ISA 7.12.2:
//   A 16x4 : lanes0-15 hold K=0(v0),K=1(v1); lanes16-31 hold K=2,K=3
//   B 4x16 : v0 = rows {0 | 2}, v1 = rows {1 | 3} split across half-waves
//   C 16x16: VGPR r -> row r (lanes0-15) and row r+8 (lanes16-31)
// ---------------------------------------------------------------------------
template <int CinT, int CoutT>
__global__ __launch_bounds__(256) void wmma_gemm_kernel(
    const float* __restrict__ Wmat,   // (CoutT, CinT) row-major
    const float* __restrict__ X,      // (B, CinT, N)
    float* __restrict__ Y)            // (B, CoutT, N)
{
    const int lane = threadIdx.x & 31;
    const int wave = threadIdx.x >> 5;      // 0..7
    const int half = lane >> 4;             // 0 or 1
    const int l16  = lane & 15;

    // 16-wide column tile (columns are (b,n) pairs; N % 16 == 0 so one batch)
    const int jt = blockIdx.x * 8 + wave;
    const int j0 = jt * 16;
    const int b  = j0 / kN;
    const int n0 = j0 % kN;
    const int obase = blockIdx.y * 64;

    const float* xb = X + (size_t)b * CinT * kN + n0;

    v8f acc0 = {}, acc1 = {}, acc2 = {}, acc3 = {};

    for (int k0 = 0; k0 < CinT; k0 += 4) {
        const int ka = k0 + 2 * half;

        // B fragment: two rows of x, 16 consecutive columns per half-wave row
        v2f bf;
        bf.x = xb[(size_t)(ka + 0) * kN + l16];
        bf.y = xb[(size_t)(ka + 1) * kN + l16];

        // A fragments: 4 output row groups, 2 consecutive K values per lane
        const float* wrow = Wmat + (size_t)(obase + l16) * CinT + ka;
        const v2f a0 = *(const v2f*)(wrow + 0 * 16 * CinT);
        const v2f a1 = *(const v2f*)(wrow + 1 * 16 * CinT);
        const v2f a2 = *(const v2f*)(wrow + 2 * 16 * CinT);
        const v2f a3 = *(const v2f*)(wrow + 3 * 16 * CinT);

        if (k0 + 8 < CinT) {
            __builtin_prefetch(&xb[(size_t)(k0 + 8 + 2 * half) * kN + l16], 0, 3);
        }

        acc0 = __builtin_amdgcn_wmma_f32_16x16x4_f32(false, a0, false, bf,
                                                     (short)0, acc0, false, false);
        acc1 = __builtin_amdgcn_wmma_f32_16x16x4_f32(false, a1, false, bf,
                                                     (short)0, acc1, false, false);
        acc2 = __builtin_amdgcn_wmma_f32_16x16x4_f32(false, a2, false, bf,
                                                     (short)0, acc2, false, false);
        acc3 = __builtin_amdgcn_wmma_f32_16x16x4_f32(false, a3, false, bf,
                                                     (short)0, acc3, false, false);
    }

    float* yb = Y + (size_t)b * CoutT * kN + n0;
    #pragma unroll
    for (int r = 0; r < 8; ++r) {
        const int orow = r + 8 * half;
        yb[(size_t)(obase + 0 * 16 + orow) * kN + l16] = acc0[r];
        yb[(size_t)(obase + 1 * 16 + orow) * kN + l16] = acc1[r];
        yb[(size_t)(obase + 2 * 16 + orow) * kN + l16] = acc2[r];
        yb[(size_t)(obase + 3 * 16 + orow) * kN + l16] = acc3[r];
    }
}

// ---------------------------------------------------------------------------
// Kernel 4: training-mode BN statistics (mean, rsqrt(var+eps)) per channel.
// One block per channel, reduces over B*N elements.
// ---------------------------------------------------------------------------
template <int C>
__global__ __launch_bounds__(256) void bn_stats_kernel(
    const float* __restrict__ Y,   // (B, C, N)
    float2* __restrict__ stats)    // (C) -> {mean, inv_std}
{
    __shared__ float ssum[256];
    __shared__ float ssq[256];

    const int c   = blockIdx.x;
    const int tid = threadIdx.x;

    float s = 0.0f, q = 0.0f;
    for (int j = tid; j < kB * kN; j += 256) {
        const int b = j >> 13;          // j / 8192
        const int n = j & (kN - 1);
        const float v = Y[((size_t)b * C + c) * kN + n];
        s += v;
        q += v * v;
    }
    ssum[tid] = s;
    ssq[tid]  = q;
    __syncthreads();

    for (int stride = 128; stride > 0; stride >>= 1) {
        if (tid < stride) {
            ssum[tid] += ssum[tid + stride];
            ssq[tid]  += ssq[tid + stride];
        }
        __syncthreads();
    }

    if (tid == 0) {
        const float inv_n = 1.0f / (float)(kB * kN);
        const float mean  = ssum[0] * inv_n;
        const float var   = ssq[0] * inv_n - mean * mean;
        float2 st;
        st.x = mean;
        st.y = rsqrtf(var + kEpsBN);
        stats[c] = st;
    }
}

// ---------------------------------------------------------------------------
// Kernel 5: apply BN (gamma/beta) + ReLU, elementwise; in-place safe.
// ---------------------------------------------------------------------------
template <int C>
__global__ __launch_bounds__(256) void bn_relu_kernel(
    const float* Yin, float* Yout,
    const float2* __restrict__ stats,
    const float* __restrict__ gamma,
    const float* __restrict__ beta)
{
    const size_t i = (size_t)blockIdx.x * blockDim.x + threadIdx.x;
    const int c = (int)((i / kN) % C);
    const float2 st = stats[c];
    const float v = (Yin[i] - st.x) * st.y * gamma[c] + beta[c];
    Yout[i] = fmaxf(v, 0.0f);
}

// ---------------------------------------------------------------------------
extern "C" void kernel_launch(void* const* d_in, const int* in_sizes, int n_in,
                              void* d_out, int out_size, void* d_ws, size_t ws_size,
                              hipStream_t stream) {
    const float* unknown      = (const float*)d_in[0];   // (B,N,3)
    const float* known        = (const float*)d_in[1];   // (B,M,3)
    const float* unknow_feats = (const float*)d_in[2];   // (B,C1,N)
    const float* known_feats  = (const float*)d_in[3];   // (B,C2,M)
    const float* W1           = (const float*)d_in[4];   // (H1, Cin)
    const float* g1           = (const float*)d_in[5];
    const float* b1           = (const float*)d_in[6];
    const float* W2           = (const float*)d_in[7];   // (H2, H1)
    const float* g2           = (const float*)d_in[8];
    const float* b2           = (const float*)d_in[9];
    float* out = (float*)d_out;                          // (B,H2,N)

    // Workspace carve-out (all buffers fully overwritten each call).
    char* ws = (char*)d_ws;
    size_t cur = 0;
    auto carve = [&](size_t bytes) -> void* {
        void* p = ws + cur;
        cur = (cur + bytes + 255) & ~(size_t)255;
        return p;
    };
    int*    idx   = (int*)   carve((size_t)kB * kN * 3 * sizeof(int));
    float*  wgt   = (float*) carve((size_t)kB * kN * 3 * sizeof(float));
    float*  xbuf  = (float*) carve((size_t)kB * kCin * kN * sizeof(float));  // ~50 MB
    float*  y1    = (float*) carve((size_t)kB * kH1 * kN * sizeof(float));   // ~34 MB
    float2* st1   = (float2*)carve((size_t)kH1 * sizeof(float2));
    float2* st2   = (float2*)carve((size_t)kH2 * sizeof(float2));
    (void)ws_size; (void)in_sizes; (void)n_in; (void)out_size;

    const dim3 blk256(256);

    // 1) 3-NN + weights
    knn_kernel<<<dim3(kN / 256, kB), blk256, 0, stream>>>(unknown, known, idx, wgt);

    // 2) interpolate + concat -> x (B, 384, N)
    build_x_kernel<<<dim3(kN / 256, kB), blk256, 0, stream>>>(
        known_feats, unknow_feats, idx, wgt, xbuf);

    // 3) layer1 GEMM (WMMA fp32): y1 = W1 * x
    wmma_gemm_kernel<kCin, kH1><<<dim3((kB * kN / 16) / 8, kH1 / 64), blk256, 0, stream>>>(
        W1, xbuf, y1);

    // 4) BN stats + BN/ReLU (in place on y1)
    bn_stats_kernel<kH1><<<dim3(kH1), blk256, 0, stream>>>(y1, st1);
    bn_relu_kernel<kH1><<<dim3((kB * kH1 * kN) / 256), blk256, 0, stream>>>(
        y1, y1, st1, g1, b1);

    // 5) layer2 GEMM (WMMA fp32): out_raw = W2 * y1  (written straight to d_out)
    wmma_gemm_kernel<kH1, kH2><<<dim3((kB * kN / 16) / 8, kH2 / 64), blk256, 0, stream>>>(
        W2, y1, out);

    // 6) BN stats + BN/ReLU (in place on d_out)
    bn_stats_kernel<kH2><<<dim3(kH2), blk256, 0, stream>>>(out, st2);
    bn_relu_kernel<kH2><<<dim3((kB * kH2 * kN) / 256), blk256, 0, stream>>>(
        out, out, st2, g2, b2);
}